// KDLoss_RM_PGFI_89103391523396
// MI455X (gfx1250) — compile-verified
//
#include <hip/hip_runtime.h>
#include <cstdint>

#define NA     8400
#define BDIM   256
#define NCHUNK ((NA + BDIM - 1) / BDIM)   // 33
#define BATCH  16
#define NGT    20

typedef __attribute__((ext_vector_type(2))) float v2f;
typedef __attribute__((ext_vector_type(8))) float v8f;

// ---------------------------------------------------------------------------
// Kernel 1: decode (reg + grid)*stride / exp(reg)*stride for both nets.
// Box layout in workspace: [b][anchor] float4 (x,y,w,h), anchors ordered
// level0 (80x80, s=8) row-major, level1 (40x40, s=16), level2 (20x20, s=32).
// ---------------------------------------------------------------------------
__global__ __launch_bounds__(256) void decode_kernel(
    const float* __restrict__ st_reg0, const float* __restrict__ st_reg1,
    const float* __restrict__ st_reg2, const float* __restrict__ te_reg0,
    const float* __restrict__ te_reg1, const float* __restrict__ te_reg2,
    float4* __restrict__ st_box, float4* __restrict__ te_box) {
  int idx = blockIdx.x * blockDim.x + threadIdx.x;   // b*NA + a
  if (idx >= BATCH * NA) return;
  int b = idx / NA;
  int a = idx - b * NA;

  const float *sreg, *treg;
  int w, hw, loc;
  float stride;
  if (a < 6400)      { sreg = st_reg0; treg = te_reg0; w = 80; hw = 6400; stride =  8.f; loc = a; }
  else if (a < 8000) { sreg = st_reg1; treg = te_reg1; w = 40; hw = 1600; stride = 16.f; loc = a - 6400; }
  else               { sreg = st_reg2; treg = te_reg2; w = 20; hw =  400; stride = 32.f; loc = a - 8000; }
  int y = loc / w;
  int x = loc - y * w;

  const float* s = sreg + (size_t)b * 4 * hw + loc;
  const float* t = treg + (size_t)b * 4 * hw + loc;

  float4 sb, tb;
  sb.x = (s[0]        + (float)x) * stride;
  sb.y = (s[hw]       + (float)y) * stride;
  sb.z = expf(s[2*hw]) * stride;
  sb.w = expf(s[3*hw]) * stride;
  tb.x = (t[0]        + (float)x) * stride;
  tb.y = (t[hw]       + (float)y) * stride;
  tb.z = expf(t[2*hw]) * stride;
  tb.w = expf(t[3*hw]) * stride;

  st_box[idx] = sb;
  te_box[idx] = tb;
}

// ---------------------------------------------------------------------------
// IoU of pred box p (cx,cy,w,h) vs gt edges (gl,gt_,gr,gb), gt area area_g.
// ---------------------------------------------------------------------------
__device__ __forceinline__ float iou_one(float4 p, float gl, float gt_,
                                         float gr, float gb, float area_g) {
  float hw = 0.5f * p.z, hh = 0.5f * p.w;
  float tlx = fmaxf(gl,  p.x - hw);
  float tly = fmaxf(gt_, p.y - hh);
  float brx = fminf(gr,  p.x + hw);
  float bry = fminf(gb,  p.y + hh);
  float ai  = ((tlx < brx) && (tly < bry)) ? (brx - tlx) * (bry - tly) : 0.0f;
  return ai / (area_g + p.z * p.w - ai + 1e-16f);
}

// ---------------------------------------------------------------------------
// Kernel 2: one block per (gt g, image b). Single streaming pass over 8400
// anchors with online softmax state:
//   (M_st, S_st)       running max / rescaled sum of exp for student logits
//   (M_te, S_te, T)    same for teacher, T = sum exp(ti-M_te)*(ti - si)
// KL = T/S_te + (M_st - M_te) + (log S_st - log S_te)    (0 if no anchor hit)
// Boxes are staged into LDS with double-buffered async copies; each lane
// consumes exactly the slot it loaded -> only s_wait_asynccnt, no barriers.
// ---------------------------------------------------------------------------
__global__ __launch_bounds__(BDIM) void rm_loss_kernel(
    const float4* __restrict__ st_box, const float4* __restrict__ te_box,
    const float* __restrict__ labels, float* __restrict__ partials) {
  __shared__ float4 lst[2][BDIM];
  __shared__ float4 lte[2][BDIM];
  __shared__ float  red[BDIM / 32][5];

  const int g = blockIdx.x;
  const int b = blockIdx.y;
  const int t = threadIdx.x;

  const float* lab = labels + (size_t)(b * NGT + g) * 5;
  const float gcx = lab[1], gcy = lab[2], gw = lab[3], gh = lab[4];
  const float gl  = gcx - 0.5f * gw, gr = gcx + 0.5f * gw;
  const float gt_ = gcy - 0.5f * gh, gb = gcy + 0.5f * gh;
  const float area_g = gw * gh;

  const size_t base = (size_t)b * NA;

  auto issue = [&](int buf, int c) {
    int a = c * BDIM + t;
    if (a >= NA) a = NA - 1;                     // clamp tail lanes (data unused)
    uint64_t gs = (uint64_t)(uintptr_t)(st_box + base + a);
    uint64_t gt64 = (uint64_t)(uintptr_t)(te_box + base + a);
    unsigned ls = (unsigned)(uintptr_t)(&lst[buf][t]);   // LDS byte address (addr[31:0])
    unsigned lt = (unsigned)(uintptr_t)(&lte[buf][t]);
    asm volatile("global_load_async_to_lds_b128 %0, %1, off" :: "v"(ls), "v"(gs)   : "memory");
    asm volatile("global_load_async_to_lds_b128 %0, %1, off" :: "v"(lt), "v"(gt64) : "memory");
  };

  float Mst = -1e30f, Sst = 0.f;
  float Mte = -1e30f, Ste = 0.f, T = 0.f;

  issue(0, 0);
  for (int c = 0; c < NCHUNK; ++c) {
    const int buf = c & 1;
    if (c + 1 < NCHUNK) {
      issue(buf ^ 1, c + 1);
      asm volatile("s_wait_asynccnt 0x2" ::: "memory");   // chunk c done (2 newer in flight)
    } else {
      asm volatile("s_wait_asynccnt 0x0" ::: "memory");
    }

    const int a = c * BDIM + t;
    if (a < NA) {
      float4 sb = lst[buf][t];
      float4 tb = lte[buf][t];

      // anchor center
      int w, loc; float stride;
      if (a < 6400)      { loc = a;        w = 80; stride =  8.f; }
      else if (a < 8000) { loc = a - 6400; w = 40; stride = 16.f; }
      else               { loc = a - 8000; w = 20; stride = 32.f; }
      int yy = loc / w;
      int xx = loc - yy * w;
      float cx = ((float)xx + 0.5f) * stride;
      float cy = ((float)yy + 0.5f) * stride;

      float dmin = fminf(fminf(cx - gl, cy - gt_), fminf(gr - cx, gb - cy));
      if (dmin > 0.0f) {
        float si = iou_one(sb, gl, gt_, gr, gb, area_g);
        float ti = iou_one(tb, gl, gt_, gr, gb, area_g);
        // student online softmax state
        float nM = fmaxf(Mst, si);
        Sst = Sst * expf(Mst - nM) + expf(si - nM);
        Mst = nM;
        // teacher online softmax state + weighted term
        nM = fmaxf(Mte, ti);
        float sc = expf(Mte - nM);
        float e  = expf(ti - nM);
        Ste = Ste * sc + e;
        T   = T   * sc + e * (ti - si);
        Mte = nM;
      }
    }
  }

  // wave32 butterfly merge of softmax states
  for (int off = 16; off > 0; off >>= 1) {
    float M2 = __shfl_xor(Mst, off);
    float S2 = __shfl_xor(Sst, off);
    float nM = fmaxf(Mst, M2);
    Sst = Sst * expf(Mst - nM) + S2 * expf(M2 - nM);
    Mst = nM;

    float M2t = __shfl_xor(Mte, off);
    float S2t = __shfl_xor(Ste, off);
    float T2  = __shfl_xor(T,   off);
    nM = fmaxf(Mte, M2t);
    float s1 = expf(Mte - nM), s2 = expf(M2t - nM);
    Ste = Ste * s1 + S2t * s2;
    T   = T   * s1 + T2  * s2;
    Mte = nM;
  }
  if ((t & 31) == 0) {
    int wv = t >> 5;
    red[wv][0] = Mst; red[wv][1] = Sst;
    red[wv][2] = Mte; red[wv][3] = Ste; red[wv][4] = T;
  }
  __syncthreads();
  if (t == 0) {
    Mst = red[0][0]; Sst = red[0][1];
    Mte = red[0][2]; Ste = red[0][3]; T = red[0][4];
    for (int i = 1; i < BDIM / 32; ++i) {
      float nM = fmaxf(Mst, red[i][0]);
      Sst = Sst * expf(Mst - nM) + red[i][1] * expf(red[i][0] - nM);
      Mst = nM;
      nM = fmaxf(Mte, red[i][2]);
      float s1 = expf(Mte - nM), s2 = expf(red[i][2] - nM);
      Ste = Ste * s1 + red[i][3] * s2;
      T   = T   * s1 + red[i][4] * s2;
      Mte = nM;
    }
    float kl = (Ste > 0.0f)
                   ? (T / Ste + (Mst - Mte) + (logf(Sst) - logf(Ste)))
                   : 0.0f;
    partials[b * NGT + g] = kl;
  }
}

// ---------------------------------------------------------------------------
// Kernel 3: WMMA reduction of the 320 partials (16 rows x 20 cols).
// Five 16x4 A-tiles x all-ones 4x16 B, accumulated into one f32 16x16 C via
// chained v_wmma_f32_16x16x4_f32. With B == 1, every D entry is a row-sum of
// A, so sum(D) == 16 * sum(partials) independent of the physical VGPR layout
// (each partial enters exactly one A slot exactly once). Single wave32 block,
// EXEC all ones, deterministic accumulation order.
// ---------------------------------------------------------------------------
__global__ __launch_bounds__(32) void wmma_reduce(const float* __restrict__ partials,
                                                  float* __restrict__ out) {
  const int lane = threadIdx.x;              // 0..31, one full wave
  const int m = lane & 15;                   // matrix row
  const int kh = (lane >> 4) << 1;           // K pair: lanes 0-15 -> K0/K1, 16-31 -> K2/K3

  v2f ones; ones.x = 1.0f; ones.y = 1.0f;    // B = all ones (4x16)
  v8f c = {};

  for (int i = 0; i < 5; ++i) {              // 5 tiles cover 20 columns
    v2f a;
    a.x = partials[m * NGT + i * 4 + kh];
    a.y = partials[m * NGT + i * 4 + kh + 1];
    c = __builtin_amdgcn_wmma_f32_16x16x4_f32(
        /*neg_a=*/false, a, /*neg_b=*/false, ones,
        /*c_mod=*/(short)0, c, /*reuse_a=*/false, /*reuse_b=*/false);
  }

  float local = c[0] + c[1] + c[2] + c[3] + c[4] + c[5] + c[6] + c[7];
  for (int off = 16; off > 0; off >>= 1) local += __shfl_xor(local, off);

  if (lane == 0)
    out[0] = local * (1.0f / 16.0f) * (1.0f / (float)(BATCH * NGT));
}

// ---------------------------------------------------------------------------
extern "C" void kernel_launch(void* const* d_in, const int* in_sizes, int n_in,
                              void* d_out, int out_size, void* d_ws, size_t ws_size,
                              hipStream_t stream) {
  (void)in_sizes; (void)n_in; (void)out_size; (void)ws_size;
  // input order: for pfx in (st, te): for lvl 0..2: reg, obj, cls ; then labels
  const float* st_reg0 = (const float*)d_in[0];
  const float* st_reg1 = (const float*)d_in[3];
  const float* st_reg2 = (const float*)d_in[6];
  const float* te_reg0 = (const float*)d_in[9];
  const float* te_reg1 = (const float*)d_in[12];
  const float* te_reg2 = (const float*)d_in[15];
  const float* labels  = (const float*)d_in[18];

  float4* st_box  = (float4*)d_ws;
  float4* te_box  = st_box + (size_t)BATCH * NA;
  float* partials = (float*)(te_box + (size_t)BATCH * NA);
  float* out      = (float*)d_out;

  const int total = BATCH * NA;                       // 134400
  decode_kernel<<<(total + 255) / 256, 256, 0, stream>>>(
      st_reg0, st_reg1, st_reg2, te_reg0, te_reg1, te_reg2, st_box, te_box);
  rm_loss_kernel<<<dim3(NGT, BATCH), BDIM, 0, stream>>>(st_box, te_box, labels, partials);
  wmma_reduce<<<1, 32, 0, stream>>>(partials, out);
}